// PlaneValidator_71854802862657
// MI455X (gfx1250) — compile-verified
//
#include <hip/hip_runtime.h>
#include <math.h>

typedef __attribute__((ext_vector_type(2))) float v2f;
typedef __attribute__((ext_vector_type(4))) float v4f;
typedef __attribute__((ext_vector_type(8))) float v8f;

#define BATCH 8
#define HPL   16
#define NPTS  1024
#define TB    1024            // threads per block in pv_sde: 32 waves fills a WGP
#define SDE_THRESH 10.0f
#define ANG_THRESH 30.0f

// ---------------- Kernel A: per-batch centroid (deterministic wave reduction) ----------
__global__ void pv_centroid(const float* __restrict__ sp, float* __restrict__ cm) {
    int b = blockIdx.x;
    int lane = threadIdx.x;
    float sx = 0.f, sy = 0.f, sz = 0.f;
    for (int p = lane; p < NPTS; p += 32) {
        const float* q = sp + (size_t)b * NPTS * 3 + 3 * p;
        sx += q[0]; sy += q[1]; sz += q[2];
    }
    #pragma unroll
    for (int m = 16; m >= 1; m >>= 1) {
        sx += __shfl_xor(sx, m);
        sy += __shfl_xor(sy, m);
        sz += __shfl_xor(sz, m);
    }
    if (lane == 0) {
        cm[b * 3 + 0] = sx * (1.0f / NPTS);
        cm[b * 3 + 1] = sy * (1.0f / NPTS);
        cm[b * 3 + 2] = sz * (1.0f / NPTS);
    }
}

// ---------------- Kernel B: per-(b,h) symmetric-distance error via f32 WMMA ------------
// d2[i,j] = rr[i] + ss[j] - 2*refl_i.s_j as one V_WMMA_F32_16X16X4_F32 per 16x16 tile:
//   A = [-2rx, -2ry, -2rz, rr] (16x4), B = [sx, sy, sz, 1]^T (4x16), C = ss[j] broadcast.
// LDS layouts make the lane-half (K=0,1 vs K=2,3) selection pure address math:
//   s_bq[q] = q<1024 ? {sx,sy,ss,0} : {sz,1,ss,0}   -> one ds_load per tile (B + C)
//   s_ap[q] = q<1024 ? {-2rx,-2ry}  : {-2rz,rr}     -> one ds_load_b64 per i-tile (A)
// Row mins: per-lane v8f accumulator + 16-lane butterfly. Col mins: ds_min_num_f32
// (exact, order-independent -> deterministic). 32 waves/workgroup = 8 waves per SIMD32
// so the DS-counter latency chain (tile load ordered behind the ds_min atomic) is hidden.
__launch_bounds__(TB)
__global__ void pv_sde(const float* __restrict__ yp, const float* __restrict__ sp,
                       float* __restrict__ sde_out) {
    __shared__ __align__(16) float s_bq[2 * NPTS * 4];   // 32 KB: B quads {bx,by,ss,0}
    __shared__ __align__(16) float s_ap[2 * NPTS * 2];   // 16 KB: A pairs
    __shared__ float s_colmin[NPTS];                     //  4 KB
    __shared__ float s_rowsum[64];

    const int bh  = blockIdx.x;           // b*16 + h
    const int b   = bh >> 4;
    const int tid = threadIdx.x;

    // plane params (uniform)
    float nx = yp[bh * 4 + 0], ny = yp[bh * 4 + 1], nz = yp[bh * 4 + 2];
    float pd = yp[bh * 4 + 3];
    float inv = 1.0f / sqrtf(nx * nx + ny * ny + nz * nz);
    nx *= inv; ny *= inv; nz *= inv;

    const float* spb = sp + (size_t)b * NPTS * 3;
    for (int j = tid; j < NPTS; j += TB) {
        float x = spb[3 * j + 0], y = spb[3 * j + 1], z = spb[3 * j + 2];
        float ssv = x * x + y * y + z * z;
        float proj = x * nx + y * ny + z * nz + pd;
        float rx = x - 2.0f * proj * nx;
        float ry = y - 2.0f * proj * ny;
        float rz = z - 2.0f * proj * nz;
        float rr = rx * rx + ry * ry + rz * rz;
        v4f q0; q0.x = x; q0.y = y;    q0.z = ssv; q0.w = 0.0f;
        v4f q1; q1.x = z; q1.y = 1.0f; q1.z = ssv; q1.w = 0.0f;
        *(v4f*)&s_bq[4 * j]          = q0;       // lanes 0-15: K=0,1 + ss
        *(v4f*)&s_bq[4 * (j + NPTS)] = q1;       // lanes 16-31: K=2,3 + ss
        v2f a0; a0.x = -2.0f * rx; a0.y = -2.0f * ry;
        v2f a1; a1.x = -2.0f * rz; a1.y = rr;
        *(v2f*)&s_ap[2 * j]          = a0;
        *(v2f*)&s_ap[2 * (j + NPTS)] = a1;
        s_colmin[j] = __builtin_inff();
    }
    __syncthreads();

    const int wave = tid >> 5;            // 0..31
    const int lane = tid & 31;
    const int lo   = lane & 15;
    const int hi   = lane >> 4;
    const int half = hi << 10;            // 0 or 1024 (entry offset for K=2,3 halves)

    for (int it = 0; it < 2; ++it) {      // 32 waves x 2 i-tiles = 64 i-tiles
        const int itile = wave * 2 + it;
        const int M = itile * 16 + lo;
        const v2f A = *(const v2f*)&s_ap[(half + M) * 2];

        v8f rowAcc;
        #pragma unroll
        for (int k = 0; k < 8; ++k) rowAcc[k] = __builtin_inff();

        for (int jt = 0; jt < 64; ++jt) {
            const v4f q = *(const v4f*)&s_bq[(half + jt * 16 + lo) * 4];
            v2f Bm; Bm.x = q.x; Bm.y = q.y;
            v8f C;
            #pragma unroll
            for (int k = 0; k < 8; ++k) C[k] = q.z;

            v8f D = __builtin_amdgcn_wmma_f32_16x16x4_f32(
                false, A, false, Bm, (short)0, C, false, false);

            float cmin = D[0];
            #pragma unroll
            for (int k = 0; k < 8; ++k) {
                rowAcc[k] = fminf(rowAcc[k], D[k]);
                if (k) cmin = fminf(cmin, D[k]);
            }
            // exact (order-independent) LDS float atomic min across waves
            unsigned off = (unsigned)(size_t)&s_colmin[jt * 16 + lo];
            asm volatile("ds_min_num_f32 %0, %1" :: "v"(off), "v"(cmin) : "memory");
        }

        // row-min: reduce across the 16 lanes of each half (masks 1..8 stay in-group)
        #pragma unroll
        for (int m = 1; m <= 8; m <<= 1) {
            #pragma unroll
            for (int k = 0; k < 8; ++k)
                rowAcc[k] = fminf(rowAcc[k], __shfl_xor(rowAcc[k], m));
        }
        float rsum = 0.f;
        #pragma unroll
        for (int k = 0; k < 8; ++k) rsum += rowAcc[k];   // 8 rows of this half
        float other = __shfl_xor(rsum, 16);              // other half's 8 rows
        if (lane == 0) s_rowsum[itile] = rsum + other;
    }

    asm volatile("s_wait_dscnt 0" ::: "memory");  // drain inline-asm DS atomics
    __syncthreads();

    if (wave == 0) {
        float cs = 0.f;
        for (int j = lane; j < NPTS; j += 32) cs += s_colmin[j];
        #pragma unroll
        for (int m = 16; m >= 1; m >>= 1) cs += __shfl_xor(cs, m);
        if (lane == 0) {
            float rs = 0.f;
            for (int t = 0; t < 64; ++t) rs += s_rowsum[t];
            sde_out[bh] = (rs + cs) * (1.0f / NPTS);
        }
    }
}

// ---------------- Kernel C: confidence, suppression, stable sort, emit -----------------
__global__ void pv_final(const float* __restrict__ yp, const float* __restrict__ ws_cm,
                         const float* __restrict__ ws_sde, float* __restrict__ out) {
    __shared__ float s_nx[HPL], s_ny[HPL], s_nz[HPL], s_d[HPL];
    __shared__ float s_sde[HPL], s_conf[HPL], s_key[HPL];
    __shared__ int   s_keep[HPL];
    const int b = blockIdx.x;
    const int h = threadIdx.x;

    if (h < HPL) {
        const float* q = yp + ((size_t)b * HPL + h) * 4;
        float nx = q[0], ny = q[1], nz = q[2];
        float inv = 1.0f / sqrtf(nx * nx + ny * ny + nz * nz);
        s_nx[h] = nx * inv; s_ny[h] = ny * inv; s_nz[h] = nz * inv;
        s_d[h]  = q[3];
        s_sde[h] = ws_sde[b * HPL + h];
    }
    __syncthreads();

    if (h < HPL) {
        float mn = __builtin_inff(), mx = -__builtin_inff();
        for (int g = 0; g < HPL; ++g) { mn = fminf(mn, s_sde[g]); mx = fmaxf(mx, s_sde[g]); }
        float sde  = s_sde[h];
        float conf = 1.0f - (sde - mn) / fabsf(mx - mn);
        bool valid = sde <= SDE_THRESH;
        bool sup = false;
        float nx = s_nx[h], ny = s_ny[h], nz = s_nz[h];
        for (int g = 0; g < HPL; ++g) {
            if (g == h) continue;
            float c = nx * s_nx[g] + ny * s_ny[g] + nz * s_nz[g];
            c = fminf(1.0f, fmaxf(-1.0f, c));
            float ang = acosf(c) * 57.29577951308232f;
            bool close = (ang < ANG_THRESH) || ((180.0f - ang) < ANG_THRESH);
            if (close && (s_sde[g] <= SDE_THRESH) && (sde >= s_sde[g])) sup = true;
        }
        bool keep = valid && !sup;
        s_conf[h] = conf;
        s_keep[h] = keep ? 1 : 0;
        s_key[h]  = keep ? conf : -__builtin_inff();
    }
    __syncthreads();

    if (h < HPL) {
        float key = s_key[h];
        int rank = 0;
        for (int g = 0; g < HPL; ++g) {
            float kg = s_key[g];
            if (kg > key || (kg == key && g < h)) rank++;   // stable descending
        }
        float* o = out + ((size_t)b * HPL + rank) * 8;
        if (s_keep[h]) {
            float nx = s_nx[h], ny = s_ny[h], nz = s_nz[h];
            float cx = ws_cm[b * 3 + 0], cy = ws_cm[b * 3 + 1], cz = ws_cm[b * 3 + 2];
            float t = nx * cx + ny * cy + nz * cz + s_d[h];
            o[0] = nx; o[1] = ny; o[2] = nz;
            o[3] = cx - t * nx; o[4] = cy - t * ny; o[5] = cz - t * nz;
            o[6] = s_conf[h]; o[7] = s_sde[h];
        } else {
            #pragma unroll
            for (int c = 0; c < 8; ++c) o[c] = 0.0f;
        }
    }
}

extern "C" void kernel_launch(void* const* d_in, const int* in_sizes, int n_in,
                              void* d_out, int out_size, void* d_ws, size_t ws_size,
                              hipStream_t stream) {
    const float* yp = (const float*)d_in[0];   // y_pred (8,16,4)
    const float* sp = (const float*)d_in[1];   // sample_points (8,1024,3)
    float* out = (float*)d_out;                // (8,16,8)
    float* cm  = (float*)d_ws;                 // 24 floats
    float* sde = cm + 32;                      // 128 floats

    pv_centroid<<<BATCH, 32, 0, stream>>>(sp, cm);
    pv_sde<<<BATCH * HPL, TB, 0, stream>>>(yp, sp, sde);
    pv_final<<<BATCH, 32, 0, stream>>>(yp, cm, sde, out);
}